// AtomAttentionEncoder_20306605375694
// MI455X (gfx1250) — compile-verified
//
#include <hip/hip_runtime.h>
#include <hip/hip_bf16.h>
#include <cstddef>

// ---------------------------------------------------------------------------
// AtomAttentionEncoder forward for MI455X (gfx1250, wave32, WMMA).
// fp32 end-to-end using V_WMMA_F32_16X16X4_F32.
// ---------------------------------------------------------------------------

typedef float v2f __attribute__((ext_vector_type(2)));
typedef float v8f __attribute__((ext_vector_type(8)));

__device__ __forceinline__ v8f wmma4(v2f a, v2f b, v8f c) {
  // D(16x16,f32) = A(16x4,f32) x B(4x16,f32) + C
  return __builtin_amdgcn_wmma_f32_16x16x4_f32(false, a, false, b, (short)0, c,
                                               false, false);
}

#define NATOM 4096
#define NTOK  1024
#define KWIN  128
#define WQ    32
#define HK    128
#define CS    128   // ATOM_S
#define CZ    16    // ATOM_Z

// ---------------------------------------------------------------------------
// Generic LDS-tiled WMMA GEMM:  Y[M,Nout] = act( (relu?)X[M,Kd] @ W[Nout,Kd]^T + bias )
// block = 256 thr (8 waves) computes 32 rows x 64 cols; grid = (ceil(N/64), M/32)
// ---------------------------------------------------------------------------
__global__ void gemm_ws(const float* __restrict__ X, const float* __restrict__ W,
                        const float* __restrict__ bias, float* __restrict__ Y,
                        int M, int Kd, int Nout, int act, int in_relu) {
  __shared__ float As[32][33];
  __shared__ float Bs[64][33];
  const int tid  = threadIdx.x;
  const int wave = tid >> 5, lane = tid & 31;
  const int m0b = blockIdx.y * 32;
  const int n0b = blockIdx.x * 64;
  const int mt = wave >> 2, nt = wave & 3;
  const int mrow = lane & 15;
  const int ks = (lane >> 4) << 1;   // K sub-offset per A/B lane layout
  v8f acc = {};
  for (int k0 = 0; k0 < Kd; k0 += 32) {
    for (int i = tid; i < 32 * 32; i += 256) {
      int r = i >> 5, cc = i & 31;
      int gk = k0 + cc, gm = m0b + r;
      float v = (gk < Kd && gm < M) ? X[(size_t)gm * Kd + gk] : 0.f;
      if (in_relu) v = fmaxf(v, 0.f);
      As[r][cc] = v;
    }
    for (int i = tid; i < 64 * 32; i += 256) {
      int r = i >> 5, cc = i & 31;
      int gk = k0 + cc, gn = n0b + r;
      Bs[r][cc] = (gk < Kd && gn < Nout) ? W[(size_t)gn * Kd + gk] : 0.f;
    }
    __syncthreads();
#pragma unroll
    for (int kk = 0; kk < 32; kk += 4) {
      v2f a, b;
      a.x = As[mt * 16 + mrow][kk + ks];
      a.y = As[mt * 16 + mrow][kk + ks + 1];
      b.x = Bs[nt * 16 + mrow][kk + ks];
      b.y = Bs[nt * 16 + mrow][kk + ks + 1];
      acc = wmma4(a, b, acc);
    }
    __syncthreads();
  }
  const int ncol = n0b + nt * 16 + mrow;
  if (ncol < Nout) {
    const float bv = bias ? bias[ncol] : 0.f;
    const int rbase = m0b + mt * 16 + ((lane >> 4) << 3);
#pragma unroll
    for (int i = 0; i < 8; ++i) {
      int row = rbase + i;
      if (row < M) {
        float v = acc[i] + bv;
        if (act == 1) v = fmaxf(v, 0.f);
        Y[(size_t)row * Nout + ncol] = v;
      }
    }
  }
}

// ---------------------------------------------------------------------------
// LayerNorm over rows (one wave per row). scale/bias optional.
// ---------------------------------------------------------------------------
__global__ void ln_rows(const float* __restrict__ X, float* __restrict__ Y,
                        const float* __restrict__ scale,
                        const float* __restrict__ bias, int M, int D) {
  int wv = (blockIdx.x * blockDim.x + threadIdx.x) >> 5;
  int lane = threadIdx.x & 31;
  if (wv >= M) return;
  const float* xr = X + (size_t)wv * D;
  float s = 0.f, ss = 0.f;
  for (int d = lane; d < D; d += 32) { float v = xr[d]; s += v; ss += v * v; }
  for (int off = 16; off; off >>= 1) {
    s  += __shfl_xor(s, off, 32);
    ss += __shfl_xor(ss, off, 32);
  }
  float mean = s / (float)D;
  float var  = ss / (float)D - mean * mean;
  float inv  = rsqrtf(var + 1e-5f);
  float* yr = Y + (size_t)wv * D;
  for (int d = lane; d < D; d += 32) {
    float v = (xr[d] - mean) * inv;
    if (scale) v = v * scale[d] + bias[d];
    yr[d] = v;
  }
}

// ---------------------------------------------------------------------------
// Elementwise: Y = sigmoid(G)*Bm (+ Add)
// ---------------------------------------------------------------------------
__global__ void ew_sigmul_add(const float* __restrict__ G,
                              const float* __restrict__ Bm,
                              const float* __restrict__ Add,
                              float* __restrict__ Y, int n) {
  int i = blockIdx.x * blockDim.x + threadIdx.x;
  if (i >= n) return;
  float g = 1.f / (1.f + __expf(-G[i]));
  float v = g * Bm[i];
  if (Add) v += Add[i];
  Y[i] = v;
}

__global__ void swiglu_kernel(const float* __restrict__ H2, float* __restrict__ T,
                              int M) {  // H2: M x 512 -> T: M x 256
  int i = blockIdx.x * blockDim.x + threadIdx.x;
  if (i >= M * 256) return;
  int r = i >> 8, c = i & 255;
  float a = H2[(size_t)r * 512 + c];
  float b = H2[(size_t)r * 512 + 256 + c];
  float si = a / (1.f + __expf(-a));
  T[i] = si * b;
}

__global__ void zero_kernel(float* __restrict__ p, int n) {
  int i = blockIdx.x * blockDim.x + threadIdx.x;
  if (i < n) p[i] = 0.f;
}

// ---------------------------------------------------------------------------
// Build atom_feats = [pos(3), charge, pad, element(128), chars(256)] : N x 389
// ---------------------------------------------------------------------------
__global__ void build_feats(const float* __restrict__ pos,
                            const float* __restrict__ charge,
                            const float* __restrict__ pad,
                            const float* __restrict__ elem,
                            const float* __restrict__ chars,
                            float* __restrict__ feats) {
  int i = blockIdx.x * blockDim.x + threadIdx.x;
  if (i >= NATOM * 389) return;
  int n = i / 389, d = i - n * 389;
  float v;
  if (d < 3) v = pos[n * 3 + d];
  else if (d == 3) v = charge[n];
  else if (d == 4) v = pad[n];
  else if (d < 133) v = elem[(size_t)n * 128 + (d - 5)];
  else v = chars[(size_t)n * 256 + (d - 133)];
  feats[i] = v;
}

// q = c0 + r_in @ W_r2q^T  (only first 3 of 10 input cols are nonzero)
__global__ void init_q(const float* __restrict__ c, const float* __restrict__ r,
                       const float* __restrict__ Wr2q, float* __restrict__ q) {
  int i = blockIdx.x * blockDim.x + threadIdx.x;
  if (i >= NATOM * CS) return;
  int n = i >> 7, d = i & 127;
  const float* wr = Wr2q + d * 10;
  q[i] = c[i] + r[n * 3 + 0] * wr[0] + r[n * 3 + 1] * wr[1] + r[n * 3 + 2] * wr[2];
}

// tok_idx = argmax_t A2T[n,t]; valid = sum_t A2T[n,t]  (one wave per atom)
__global__ void tok_kernel(const float* __restrict__ A2T, int* __restrict__ tok,
                           float* __restrict__ valid) {
  int wv = (blockIdx.x * blockDim.x + threadIdx.x) >> 5;
  int lane = threadIdx.x & 31;
  if (wv >= NATOM) return;
  const float* row = A2T + (size_t)wv * NTOK;
  float best = -3.0e38f; int bidx = 0; float s = 0.f;
  for (int t = lane; t < NTOK; t += 32) {
    float v = row[t]; s += v;
    if (v > best) { best = v; bidx = t; }
  }
  for (int off = 16; off; off >>= 1) {
    float ob = __shfl_xor(best, off, 32);
    int   oi = __shfl_xor(bidx, off, 32);
    s += __shfl_xor(s, off, 32);
    if (ob > best || (ob == best && oi < bidx)) { best = ob; bidx = oi; }
  }
  if (lane == 0) { tok[wv] = bidx; valid[wv] = s; }
}

// c += valid[n] * s2c[tok[n], :]
__global__ void inject_c(float* __restrict__ c, const float* __restrict__ s2c,
                         const int* __restrict__ tok, const float* __restrict__ valid) {
  int i = blockIdx.x * blockDim.x + threadIdx.x;
  if (i >= NATOM * CS) return;
  int n = i >> 7, d = i & 127;
  c[i] += valid[n] * s2c[(size_t)tok[n] * CS + d];
}

// ---------------------------------------------------------------------------
// zp on the window footprint only: zp_win[k][i][jj][:] = LN(z[8k+i, 8k-12+jj]) @ Wz2p^T
// one wave per (k,i,jj); 32768 waves total. Reads only ~16MB of z (vs 512MB dense).
// ---------------------------------------------------------------------------
__global__ void zp_window(const float* __restrict__ z, const float* __restrict__ zs,
                          const float* __restrict__ zb, const float* __restrict__ Wz2p,
                          float* __restrict__ zp_win) {
  __shared__ float Wsm[16 * 128];
  __shared__ float scl[128], bia[128];
  for (int i = threadIdx.x; i < 16 * 128; i += 256) Wsm[i] = Wz2p[i];
  for (int i = threadIdx.x; i < 128; i += 256) { scl[i] = zs[i]; bia[i] = zb[i]; }
  __syncthreads();
  int gw = (blockIdx.x * blockDim.x + threadIdx.x) >> 5;
  int lane = threadIdx.x & 31;
  if (gw >= KWIN * 8 * 32) return;
  int k = gw >> 8, rem = gw & 255;
  int ti = rem >> 5, jj = rem & 31;
  int t1 = 8 * k + ti;
  int t2 = 8 * k - 12 + jj;
  float out[16];
  if (t2 < 0 || t2 >= NTOK) {
    for (int o = 0; o < 16; ++o) out[o] = 0.f;
  } else {
    const float* zr = z + ((size_t)t1 * NTOK + t2) * 128;
    float x[4], y[4]; float s = 0.f, ss = 0.f;
#pragma unroll
    for (int u = 0; u < 4; ++u) { x[u] = zr[lane * 4 + u]; s += x[u]; ss += x[u] * x[u]; }
    for (int off = 16; off; off >>= 1) {
      s += __shfl_xor(s, off, 32); ss += __shfl_xor(ss, off, 32);
    }
    float mean = s * (1.f / 128.f);
    float var  = ss * (1.f / 128.f) - mean * mean;
    float inv  = rsqrtf(var + 1e-5f);
#pragma unroll
    for (int u = 0; u < 4; ++u) {
      int d = lane * 4 + u;
      y[u] = (x[u] - mean) * inv * scl[d] + bia[d];
    }
#pragma unroll
    for (int o = 0; o < 16; ++o) {
      float p = 0.f;
#pragma unroll
      for (int u = 0; u < 4; ++u) p += y[u] * Wsm[o * 128 + lane * 4 + u];
      for (int off = 16; off; off >>= 1) p += __shfl_xor(p, off, 32);
      out[o] = p;
    }
  }
  if (lane < 16) zp_win[(size_t)gw * 16 + lane] = out[lane];
}

// ---------------------------------------------------------------------------
// Pair tensor assembly + 16-wide pair MLP.  block = one (k,w) query atom,
// thread j = one key (keys of window k are atoms [32k-48, 32k+80), zero padded).
// Writes final p (also a model output) into d_out's p region.
// ---------------------------------------------------------------------------
__global__ void pair_assemble(const float* __restrict__ pos, const float* __restrict__ pad,
                              const int* __restrict__ uid, const float* __restrict__ valid,
                              const float* __restrict__ zp_win,
                              const float* __restrict__ cq16, const float* __restrict__ ck16,
                              const float* __restrict__ Wpp, const float* __restrict__ Wpd,
                              const float* __restrict__ Wpm, const float* __restrict__ W1,
                              const float* __restrict__ W2, const float* __restrict__ W3,
                              float* __restrict__ p_out) {
  __shared__ float sW1[256], sW2[256], sW3[256], sWpp[48], sWpd[16], sWpm[16];
  for (int i = threadIdx.x; i < 256; i += 128) {
    sW1[i] = W1[i]; sW2[i] = W2[i]; sW3[i] = W3[i];
  }
  if (threadIdx.x < 48) sWpp[threadIdx.x] = Wpp[threadIdx.x];
  if (threadIdx.x < 16) { sWpd[threadIdx.x] = Wpd[threadIdx.x]; sWpm[threadIdx.x] = Wpm[threadIdx.x]; }
  __syncthreads();
  const int n = blockIdx.x;           // query atom
  const int k = n >> 5, w = n & 31;
  const int j = threadIdx.x;          // key slot 0..127
  const int kb = 32 * k - 48 + j;
  const bool inr = (kb >= 0 && kb < NATOM);
  float pq0 = pos[n * 3], pq1 = pos[n * 3 + 1], pq2 = pos[n * 3 + 2];
  float pk0 = inr ? pos[kb * 3] : 0.f;
  float pk1 = inr ? pos[kb * 3 + 1] : 0.f;
  float pk2 = inr ? pos[kb * 3 + 2] : 0.f;
  float d0 = pk0 - pq0, d1 = pk1 - pq1, d2 = pk2 - pq2;
  float dn = 1.f / (1.f + d0 * d0 + d1 * d1 + d2 * d2);
  bool mq = pad[n] > 0.5f;
  bool mk = inr && (pad[kb] > 0.5f);
  int uq = uid[n];
  int uk = inr ? uid[kb] : 0;
  float v  = (mq && mk && (uq == uk)) ? 1.f : 0.f;
  float vq = valid[n];
  float vk = inr ? valid[kb] : 0.f;
  const float* zwp = zp_win + (((size_t)k * 8 + (w >> 2)) * 32 + (j >> 2)) * 16;
  float pch[16];
#pragma unroll
  for (int o = 0; o < 16; ++o) {
    float t = d0 * sWpp[o * 3] + d1 * sWpp[o * 3 + 1] + d2 * sWpp[o * 3 + 2];
    t += dn * sWpd[o];
    t += v * sWpm[o];
    t *= v;
    t += zwp[o] * vq * vk;
    t += cq16[(size_t)n * 16 + o];
    if (inr) t += ck16[(size_t)kb * 16 + o];
    pch[o] = t;
  }
  float t1[16], t2[16];
#pragma unroll
  for (int o = 0; o < 16; ++o) {
    float a = 0.f;
#pragma unroll
    for (int u = 0; u < 16; ++u) a += fmaxf(pch[u], 0.f) * sW1[o * 16 + u];
    t1[o] = a;
  }
#pragma unroll
  for (int o = 0; o < 16; ++o) {
    float a = 0.f;
#pragma unroll
    for (int u = 0; u < 16; ++u) a += fmaxf(t1[u], 0.f) * sW2[o * 16 + u];
    t2[o] = a;
  }
  float* dst = p_out + ((size_t)n * 128 + j) * 16;
#pragma unroll
  for (int o = 0; o < 16; ++o) {
    float a = 0.f;
#pragma unroll
    for (int u = 0; u < 16; ++u) a += fmaxf(t2[u], 0.f) * sW3[o * 16 + u];
    dst[o] = pch[o] + a;
  }
}

// pb[h][n][j] = LN(p[n,j,:]) . Wb_pair[h]
__global__ void pb_kernel(const float* __restrict__ p, const float* __restrict__ lnp_s,
                          const float* __restrict__ lnp_b, const float* __restrict__ Wb,
                          float* __restrict__ pbbuf) {
  int idx = blockIdx.x * blockDim.x + threadIdx.x;  // N*128
  if (idx >= NATOM * HK) return;
  const float* pr = p + (size_t)idx * 16;
  float s = 0.f, ss = 0.f;
#pragma unroll
  for (int o = 0; o < 16; ++o) { float x = pr[o]; s += x; ss += x * x; }
  float mean = s * (1.f / 16.f);
  float var  = ss * (1.f / 16.f) - mean * mean;
  float inv  = rsqrtf(var + 1e-5f);
  float y[16];
#pragma unroll
  for (int o = 0; o < 16; ++o) y[o] = (pr[o] - mean) * inv * lnp_s[o] + lnp_b[o];
#pragma unroll
  for (int h = 0; h < 4; ++h) {
    float a = 0.f;
#pragma unroll
    for (int o = 0; o < 16; ++o) a += y[o] * Wb[h * 16 + o];
    pbbuf[(size_t)h * (NATOM * HK) + idx] = a;
  }
}

// ---------------------------------------------------------------------------
// Windowed multi-head attention for one window k (block = 256 thr = 8 waves).
// Q: 32 rows, K/V: 128 sliding-window keys, 4 heads of dh=32. WMMA f32 16x16x4
// for both Q.K^T and attn.V;  softmax wave-cooperative.
// dynamic LDS: kmask(128) Qh(32x33) Kh(128x33) Vh(128x33) logits(32x128)
// ---------------------------------------------------------------------------
#define ATTN_LDS_FLOATS (128 + 32 * 33 + 128 * 33 + 128 * 33 + 32 * 128)
__global__ void attn_window(const float* __restrict__ BQ, const float* __restrict__ BK,
                            const float* __restrict__ BV, const float* __restrict__ pad,
                            const float* __restrict__ pbbuf, float* __restrict__ Obuf) {
  extern __shared__ float sm[];
  float* kmask = sm;
  float* Qh = sm + 128;
  float* Kh = Qh + 32 * 33;
  float* Vh = Kh + 128 * 33;
  float* lg = Vh + 128 * 33;
  const int k = blockIdx.x;
  const int tid = threadIdx.x;
  const int wave = tid >> 5, lane = tid & 31;
  const int mrow = lane & 15;
  const int ks = (lane >> 4) << 1;
  for (int j = tid; j < 128; j += 256) {
    int kb = 32 * k - 48 + j;
    kmask[j] = (kb >= 0 && kb < NATOM && pad[kb] > 0.5f) ? 1.f : 0.f;
  }
  for (int h = 0; h < 4; ++h) {
    for (int i = tid; i < 32 * 32; i += 256) {
      int w = i >> 5, d = i & 31;
      Qh[w * 33 + d] = BQ[((size_t)(32 * k + w)) * CS + h * 32 + d];
    }
    for (int i = tid; i < 128 * 32; i += 256) {
      int j = i >> 5, d = i & 31;
      int kb = 32 * k - 48 + j;
      float kv = 0.f, vv = 0.f;
      if (kb >= 0 && kb < NATOM) {
        kv = BK[(size_t)kb * CS + h * 32 + d];
        vv = BV[(size_t)kb * CS + h * 32 + d];
      }
      Kh[j * 33 + d] = kv;
      Vh[j * 33 + d] = vv;
    }
    __syncthreads();
    // logits = Q @ K^T * (dh^-0.5) + pb, masked -> lg
    for (int t = wave; t < 16; t += 8) {
      int mt = t >> 3, nt = t & 7;
      v8f acc = {};
#pragma unroll
      for (int kk = 0; kk < 32; kk += 4) {
        v2f a, b;
        a.x = Qh[(mt * 16 + mrow) * 33 + kk + ks];
        a.y = Qh[(mt * 16 + mrow) * 33 + kk + ks + 1];
        b.x = Kh[(nt * 16 + mrow) * 33 + kk + ks];
        b.y = Kh[(nt * 16 + mrow) * 33 + kk + ks + 1];
        acc = wmma4(a, b, acc);
      }
      int jcol = nt * 16 + mrow;
      float msk = kmask[jcol];
      int rbase = mt * 16 + ((lane >> 4) << 3);
#pragma unroll
      for (int i = 0; i < 8; ++i) {
        int w = rbase + i;
        float val = acc[i] * 0.17677669529663687f /* 32^-0.5 */
                  + pbbuf[(size_t)h * (NATOM * HK) + ((size_t)(32 * k + w)) * HK + jcol];
        lg[w * 128 + jcol] = (msk > 0.f) ? val : -1e9f;
      }
    }
    __syncthreads();
    // softmax rows
    for (int w = wave; w < 32; w += 8) {
      float x[4]; float mx = -3.0e38f;
#pragma unroll
      for (int u = 0; u < 4; ++u) { x[u] = lg[w * 128 + lane * 4 + u]; mx = fmaxf(mx, x[u]); }
      for (int off = 16; off; off >>= 1) mx = fmaxf(mx, __shfl_xor(mx, off, 32));
      float s = 0.f;
#pragma unroll
      for (int u = 0; u < 4; ++u) { x[u] = __expf(x[u] - mx); s += x[u]; }
      for (int off = 16; off; off >>= 1) s += __shfl_xor(s, off, 32);
      float inv = 1.f / s;
#pragma unroll
      for (int u = 0; u < 4; ++u) lg[w * 128 + lane * 4 + u] = x[u] * inv;
    }
    __syncthreads();
    // O = attn @ V  (32x32)
    if (wave < 4) {
      int mt = wave >> 1, nt = wave & 1;
      v8f acc = {};
      for (int kk = 0; kk < 128; kk += 4) {
        v2f a, b;
        a.x = lg[(mt * 16 + mrow) * 128 + kk + ks];
        a.y = lg[(mt * 16 + mrow) * 128 + kk + ks + 1];
        b.x = Vh[(kk + ks) * 33 + nt * 16 + mrow];
        b.y = Vh[(kk + ks + 1) * 33 + nt * 16 + mrow];
        acc = wmma4(a, b, acc);
      }
      int dcol = nt * 16 + mrow;
      int rbase = mt * 16 + ((lane >> 4) << 3);
#pragma unroll
      for (int i = 0; i < 8; ++i) {
        int w = rbase + i;
        Obuf[((size_t)(32 * k + w)) * CS + h * 32 + dcol] = acc[i];
      }
    }
    __syncthreads();
  }
}

// ---------------------------------------------------------------------------
// Token pooling: a[t,:] = sum_n valid[n]*(tok[n]==t)*QA[n,:],  then /= colsum+1e-6
// ---------------------------------------------------------------------------
__global__ void scatter_a(const float* __restrict__ QA, const int* __restrict__ tok,
                          const float* __restrict__ valid, float* __restrict__ a,
                          float* __restrict__ colsum) {
  int n = blockIdx.x;
  int t = tok[n];
  float vl = valid[n];
  if (threadIdx.x == 0) atomicAdd(&colsum[t], vl);
  for (int d = threadIdx.x; d < 768; d += 256)
    atomicAdd(&a[(size_t)t * 768 + d], vl * QA[(size_t)n * 768 + d]);
}

__global__ void scale_a(float* __restrict__ a, const float* __restrict__ colsum) {
  int i = blockIdx.x * blockDim.x + threadIdx.x;
  if (i >= NTOK * 768) return;
  a[i] *= 1.f / (colsum[i / 768] + 1e-6f);
}

// ---------------------------------------------------------------------------
// Host side
// ---------------------------------------------------------------------------
enum {
  IN_REF_POS = 0, IN_REF_CHARGE, IN_PAD_MASK, IN_REF_ELEMENT, IN_NAME_CHARS,
  IN_A2T, IN_S_TRUNK, IN_Z, IN_R, IN_UID,
  IN_W_ATOM_FEAT, IN_W_PAIRPOS, IN_W_PAIRDIST, IN_W_PAIRMASK,
  IN_LN_S_SCALE, IN_LN_S_BIAS, IN_W_S2C, IN_LN_Z_SCALE, IN_LN_Z_BIAS, IN_W_Z2P,
  IN_W_R2Q, IN_W_C2P_Q, IN_W_C2P_K, IN_W_PMLP1, IN_W_PMLP2, IN_W_PMLP3, IN_W_A2T,
  IN_ADA_WG, IN_ADA_BG, IN_ADA_WB, IN_WQ, IN_BQ, IN_WK, IN_WV,
  IN_LN_P_SCALE, IN_LN_P_BIAS, IN_WB_PAIR, IN_WGATE, IN_WO,
  IN_WS_OUT, IN_BS_OUT, IN_ADA2_WG, IN_ADA2_BG, IN_ADA2_WB,
  IN_WT1, IN_WT2, IN_WS_T, IN_BS_T
};

static inline void launch_gemm(const float* X, const float* W, const float* bias,
                               float* Y, int M, int Kd, int Nout, int act, int in_relu,
                               hipStream_t s) {
  dim3 grid((Nout + 63) / 64, (M + 31) / 32, 1);
  gemm_ws<<<grid, 256, 0, s>>>(X, W, bias, Y, M, Kd, Nout, act, in_relu);
}

extern "C" void kernel_launch(void* const* d_in, const int* in_sizes, int n_in,
                              void* d_out, int out_size, void* d_ws, size_t ws_size,
                              hipStream_t stream) {
  (void)in_sizes; (void)n_in; (void)out_size; (void)ws_size;
  const float* ref_pos   = (const float*)d_in[IN_REF_POS];
  const float* ref_chg   = (const float*)d_in[IN_REF_CHARGE];
  const float* pad       = (const float*)d_in[IN_PAD_MASK];
  const float* elem      = (const float*)d_in[IN_REF_ELEMENT];
  const float* chars     = (const float*)d_in[IN_NAME_CHARS];
  const float* A2T       = (const float*)d_in[IN_A2T];
  const float* s_trunk   = (const float*)d_in[IN_S_TRUNK];
  const float* z         = (const float*)d_in[IN_Z];
  const float* r         = (const float*)d_in[IN_R];
  const int*   uid       = (const int*)d_in[IN_UID];

  // output regions (a, q, c, p concatenated)
  float* a_out = (float*)d_out;                 // 1024*768
  float* q_out = a_out + 786432;                // 4096*128
  float* c_out = q_out + 524288;                // 4096*128
  float* p_out = c_out + 524288;                // 128*32*128*16

  // workspace layout (floats)
  float* ws = (float*)d_ws;
  size_t off = 0;
  float* feats  = ws + off; off += (size_t)NATOM * 389;
  float* sN     = ws + off; off += (size_t)NTOK * 384;
  float* s2c    = ws + off; off += (size_t)NTOK * CS;
  float* valid  = ws + off; off += NATOM;
  float* zp_win = ws + off; off += (size_t)KWIN * 8 * 32 * 16;
  float* cq16   = ws + off; off += (size_t)NATOM * 16;
  float* ck16   = ws + off; off += (size_t)NATOM * 16;
  float* S      = ws + off; off += (size_t)NATOM * CS;
  float* LQ     = ws + off; off += (size_t)NATOM * CS;
  float* G1     = ws + off; off += (size_t)NATOM * CS;
  float* G2     = ws + off; off += (size_t)NATOM * CS;
  float* Bn     = ws + off; off += (size_t)NATOM * CS;
  float* BQb    = ws + off; off += (size_t)NATOM * CS;
  float* BKb    = ws + off; off += (size_t)NATOM * CS;
  float* BVb    = ws + off; off += (size_t)NATOM * CS;
  float* Obuf   = ws + off; off += (size_t)NATOM * CS;
  float* H2     = ws + off; off += (size_t)NATOM * 512;
  float* Tb     = ws + off; off += (size_t)NATOM * 256;
  float* QA     = ws + off; off += (size_t)NATOM * 768;
  float* pbbuf  = ws + off; off += (size_t)4 * NATOM * HK;
  float* colsum = ws + off; off += NTOK;
  int*   tok    = (int*)(ws + off); off += NATOM;

  const int NC = NATOM * CS;  // 524288

  // 1) atom feature embed -> c ;  q = c + r @ W_r2q^T (q captured pre-injection)
  build_feats<<<(NATOM * 389 + 255) / 256, 256, 0, stream>>>(ref_pos, ref_chg, pad, elem, chars, feats);
  launch_gemm(feats, (const float*)d_in[IN_W_ATOM_FEAT], nullptr, c_out, NATOM, 389, CS, 0, 0, stream);
  init_q<<<NC / 256, 256, 0, stream>>>(c_out, r, (const float*)d_in[IN_W_R2Q], q_out);

  // 2) token index / validity ; trunk injection into c
  tok_kernel<<<NATOM / 8, 256, 0, stream>>>(A2T, tok, valid);
  ln_rows<<<NTOK / 8, 256, 0, stream>>>(s_trunk, sN,
      (const float*)d_in[IN_LN_S_SCALE], (const float*)d_in[IN_LN_S_BIAS], NTOK, 384);
  launch_gemm(sN, (const float*)d_in[IN_W_S2C], nullptr, s2c, NTOK, 384, CS, 0, 0, stream);
  inject_c<<<NC / 256, 256, 0, stream>>>(c_out, s2c, tok, valid);

  // 3) pair tensor p
  zp_window<<<KWIN * 8 * 32 / 8, 256, 0, stream>>>(z,
      (const float*)d_in[IN_LN_Z_SCALE], (const float*)d_in[IN_LN_Z_BIAS],
      (const float*)d_in[IN_W_Z2P], zp_win);
  launch_gemm(c_out, (const float*)d_in[IN_W_C2P_Q], nullptr, cq16, NATOM, CS, 16, 0, 1, stream);
  launch_gemm(c_out, (const float*)d_in[IN_W_C2P_K], nullptr, ck16, NATOM, CS, 16, 0, 1, stream);
  pair_assemble<<<NATOM, 128, 0, stream>>>(ref_pos, pad, uid, valid, zp_win, cq16, ck16,
      (const float*)d_in[IN_W_PAIRPOS], (const float*)d_in[IN_W_PAIRDIST],
      (const float*)d_in[IN_W_PAIRMASK], (const float*)d_in[IN_W_PMLP1],
      (const float*)d_in[IN_W_PMLP2], (const float*)d_in[IN_W_PMLP3], p_out);

  // 4) s = ln(c)  (constant across encoder layers)
  ln_rows<<<NATOM / 8, 256, 0, stream>>>(c_out, S, nullptr, nullptr, NATOM, CS);

  // 5) encoder layers
  for (int l = 0; l < 3; ++l) {
    const size_t oM = (size_t)l * CS * CS, oV = (size_t)l * CS;
    const float* ada_Wg  = (const float*)d_in[IN_ADA_WG]  + oM;
    const float* ada_bg  = (const float*)d_in[IN_ADA_BG]  + oV;
    const float* ada_Wb  = (const float*)d_in[IN_ADA_WB]  + oM;
    const float* Wq      = (const float*)d_in[IN_WQ]      + oM;
    const float* bq      = (const float*)d_in[IN_BQ]      + oV;
    const float* Wk      = (const float*)d_in[IN_WK]      + oM;
    const float* Wv      = (const float*)d_in[IN_WV]      + oM;
    const float* lnp_s   = (const float*)d_in[IN_LN_P_SCALE] + (size_t)l * 16;
    const float* lnp_b   = (const float*)d_in[IN_LN_P_BIAS]  + (size_t)l * 16;
    const float* Wb_pair = (const float*)d_in[IN_WB_PAIR] + (size_t)l * 64;
    const float* Wgate   = (const float*)d_in[IN_WGATE]   + oM;
    const float* Wo      = (const float*)d_in[IN_WO]      + oM;
    const float* Ws_out  = (const float*)d_in[IN_WS_OUT]  + oM;
    const float* bs_out  = (const float*)d_in[IN_BS_OUT]  + oV;
    const float* ada2_Wg = (const float*)d_in[IN_ADA2_WG] + oM;
    const float* ada2_bg = (const float*)d_in[IN_ADA2_BG] + oV;
    const float* ada2_Wb = (const float*)d_in[IN_ADA2_WB] + oM;
    const float* Wt1     = (const float*)d_in[IN_WT1]     + (size_t)l * 512 * CS;
    const float* Wt2     = (const float*)d_in[IN_WT2]     + (size_t)l * CS * 256;
    const float* Ws_t    = (const float*)d_in[IN_WS_T]    + oM;
    const float* bs_t    = (const float*)d_in[IN_BS_T]    + oV;

    // b = sigmoid(S@ada_Wg^T + ada_bg) * ln(q) + S@ada_Wb^T
    launch_gemm(S, ada_Wg, ada_bg, G1, NATOM, CS, CS, 0, 0, stream);
    launch_gemm(S, ada_Wb, nullptr, G2, NATOM, CS, CS, 0, 0, stream);
    ln_rows<<<NATOM / 8, 256, 0, stream>>>(q_out, LQ, nullptr, nullptr, NATOM, CS);
    ew_sigmul_add<<<NC / 256, 256, 0, stream>>>(G1, LQ, G2, Bn, NC);
    // attention projections
    launch_gemm(Bn, Wq, bq, BQb, NATOM, CS, CS, 0, 0, stream);
    launch_gemm(Bn, Wk, nullptr, BKb, NATOM, CS, CS, 0, 0, stream);
    launch_gemm(Bn, Wv, nullptr, BVb, NATOM, CS, CS, 0, 0, stream);
    // pair bias for this layer, then windowed attention
    pb_kernel<<<(NATOM * HK) / 256, 256, 0, stream>>>(p_out, lnp_s, lnp_b, Wb_pair, pbbuf);
    attn_window<<<KWIN, 256, ATTN_LDS_FLOATS * sizeof(float), stream>>>(
        BQb, BKb, BVb, pad, pbbuf, Obuf);
    // gated output projection + residual
    launch_gemm(Bn, Wgate, nullptr, G1, NATOM, CS, CS, 0, 0, stream);
    ew_sigmul_add<<<NC / 256, 256, 0, stream>>>(G1, Obuf, nullptr, Obuf, NC);
    launch_gemm(Obuf, Wo, nullptr, G2, NATOM, CS, CS, 0, 0, stream);
    launch_gemm(S, Ws_out, bs_out, G1, NATOM, CS, CS, 0, 0, stream);
    ew_sigmul_add<<<NC / 256, 256, 0, stream>>>(G1, G2, q_out, q_out, NC);
    // SwiGLU transition + residual
    launch_gemm(S, ada2_Wg, ada2_bg, G1, NATOM, CS, CS, 0, 0, stream);
    launch_gemm(S, ada2_Wb, nullptr, G2, NATOM, CS, CS, 0, 0, stream);
    ln_rows<<<NATOM / 8, 256, 0, stream>>>(q_out, LQ, nullptr, nullptr, NATOM, CS);
    ew_sigmul_add<<<NC / 256, 256, 0, stream>>>(G1, LQ, G2, Bn, NC);
    launch_gemm(Bn, Wt1, nullptr, H2, NATOM, CS, 512, 0, 0, stream);
    swiglu_kernel<<<(NATOM * 256) / 256, 256, 0, stream>>>(H2, Tb, NATOM);
    launch_gemm(Tb, Wt2, nullptr, G2, NATOM, 256, CS, 0, 0, stream);
    launch_gemm(S, Ws_t, bs_t, G1, NATOM, CS, CS, 0, 0, stream);
    ew_sigmul_add<<<NC / 256, 256, 0, stream>>>(G1, G2, q_out, q_out, NC);
  }

  // 6) token pooling: a = mean_over_atoms( relu(q @ W_a2t^T) )
  launch_gemm(q_out, (const float*)d_in[IN_W_A2T], nullptr, QA, NATOM, CS, 768, 1, 0, stream);
  zero_kernel<<<(NTOK * 768) / 256, 256, 0, stream>>>(a_out, NTOK * 768);
  zero_kernel<<<(NTOK + 255) / 256, 256, 0, stream>>>(colsum, NTOK);
  scatter_a<<<NATOM, 256, 0, stream>>>(QA, tok, valid, a_out, colsum);
  scale_a<<<(NTOK * 768) / 256, 256, 0, stream>>>(a_out, colsum);
}